// RNN_2181843386432
// MI455X (gfx1250) — compile-verified
//
#include <hip/hip_runtime.h>
#include <math.h>

typedef __attribute__((ext_vector_type(2))) float v2f;
typedef __attribute__((ext_vector_type(8))) float v8f;

#define DIM_     256
#define LATENT_  1024
#define BATCH_   128
#define TSTEPS_  512
#define KTOT_    (DIM_ + LATENT_)

// Padded LDS strides (dwords). 36: conflict-free wave32 gathers (36*l mod 64
// distinct for l=0..15; +2-offset upper-half lanes land in disjoint mod-4
// bank classes) and 16B-aligned float4 staging stores.
#define A_STRIDE  36   // A chunk: 64 M-rows x 32 K
#define BT_STRIDE 36   // B chunk, K-transposed: 64 N-rows x 32 K

__device__ __forceinline__ float fast_tanh(float v) {
#if __has_builtin(__builtin_amdgcn_tanhf)
  return __builtin_amdgcn_tanhf(v);
#else
  float r;
  // V_TANH_F32 is a gfx1250 TRANS op; keep 1 slot after it before use.
  asm volatile("v_tanh_f32 %0, %1\n\tv_nop" : "=v"(r) : "v"(v));
  return r;
#endif
}

// One recurrence step: h_out(128,1024) = tanh([x_t | h_prev] @ W_in + b)
// Block: 256 threads (8 waves) -> 64(M) x 64(N) output block.
// Wave w: M-tile (w&3), N-half (w>>2) -> two 16x16 f32 accumulators.
// K pipelined in chunks of 32 through double-buffered LDS.
__global__ __launch_bounds__(256) void rnn_step_kernel(
    const float* __restrict__ x,      // (B, T, DIM)
    const float* __restrict__ W,      // (DIM+LATENT, LATENT) row-major
    const float* __restrict__ bias,   // (LATENT)
    const float* __restrict__ h_prev, // (B, LATENT)  [unread when kTotal==DIM_]
    float* __restrict__ h_out,        // (B, LATENT) slab of h_hist for step t
    int t, int kTotal)
{
  __shared__ __align__(16) float Alds[2][64 * A_STRIDE];
  __shared__ __align__(16) float Btld[2][64 * BT_STRIDE];

  const int tid    = threadIdx.x;
  const int lane   = tid & 31;
  const int warp   = tid >> 5;
  const int wy     = warp & 3;        // M tile index in block (0..3)
  const int wx     = warp >> 2;       // N half in block (0..1)
  const int blockN = blockIdx.x * 64;
  const int blockM = blockIdx.y * 64;

  const int l16   = lane & 15;
  const int lhalf = (lane < 16) ? 0 : 2;   // K sub-offset held by this lane half

  // Staging geometry (per thread): A float4 slots at rows r, r+32; B at k, k+16.
  const int arow = tid >> 3;          // 0..31
  const int acol = (tid & 7) << 2;    // 0,4,...,28
  const int bkrw = tid >> 4;          // 0..15
  const int bcol = (tid & 15) << 2;   // 0..60

  const int nChunks = kTotal >> 5;

  v8f c0 = {};
  v8f c1 = {};
  float4 ar0, ar1, br0, br1;

  // ---- global load of K-chunk c into registers ----
  auto gload = [&](int c) {
    const int kc = c << 5;
    const int k  = kc + acol;
    const float* s0;
    const float* s1;
    if (k < DIM_) {   // uniform per chunk: 32 | 256
      s0 = x + ((long)(blockM + arow)      * TSTEPS_ + t) * DIM_ + k;
      s1 = x + ((long)(blockM + arow + 32) * TSTEPS_ + t) * DIM_ + k;
    } else {
      s0 = h_prev + (long)(blockM + arow)      * LATENT_ + (k - DIM_);
      s1 = h_prev + (long)(blockM + arow + 32) * LATENT_ + (k - DIM_);
    }
    ar0 = *(const float4*)s0;
    ar1 = *(const float4*)s1;
    const float* wb = W + (long)(kc + bkrw) * LATENT_ + blockN + bcol;
    br0 = *(const float4*)wb;
    br1 = *(const float4*)(wb + 16L * LATENT_);
    if (c + 2 < nChunks)             // warm L2 two chunks ahead
      __builtin_prefetch(wb + 64L * LATENT_, 0, 1);
  };

  // ---- registers -> LDS (B stored K-transposed) ----
  auto stage = [&](int buf) {
    *(float4*)&Alds[buf][arow * A_STRIDE + acol]        = ar0;
    *(float4*)&Alds[buf][(arow + 32) * A_STRIDE + acol] = ar1;
    float* bt = &Btld[buf][0];
    bt[(bcol + 0) * BT_STRIDE + bkrw] = br0.x;
    bt[(bcol + 1) * BT_STRIDE + bkrw] = br0.y;
    bt[(bcol + 2) * BT_STRIDE + bkrw] = br0.z;
    bt[(bcol + 3) * BT_STRIDE + bkrw] = br0.w;
    bt[(bcol + 0) * BT_STRIDE + bkrw + 16] = br1.x;
    bt[(bcol + 1) * BT_STRIDE + bkrw + 16] = br1.y;
    bt[(bcol + 2) * BT_STRIDE + bkrw + 16] = br1.z;
    bt[(bcol + 3) * BT_STRIDE + bkrw + 16] = br1.w;
  };

  gload(0);
  stage(0);
  __syncthreads();

  for (int c = 0; c < nChunks; ++c) {
    const int buf = c & 1;
    if (c + 1 < nChunks) gload(c + 1);   // in flight during compute

    // ---- hoist all fragments for this chunk, then 16 back-to-back WMMAs ----
    const float* Ab = &Alds[buf][(wy * 16 + l16) * A_STRIDE + lhalf];
    const float* B0 = &Btld[buf][(wx * 32 + l16) * BT_STRIDE + lhalf];
    const float* B1 = B0 + 16 * BT_STRIDE;
    v2f af[8], bf0[8], bf1[8];
#pragma unroll
    for (int s = 0; s < 8; ++s) {
      af[s]  = *(const v2f*)(Ab + 4 * s);
      bf0[s] = *(const v2f*)(B0 + 4 * s);
      bf1[s] = *(const v2f*)(B1 + 4 * s);
    }
#pragma unroll
    for (int s = 0; s < 8; ++s) {
      c0 = __builtin_amdgcn_wmma_f32_16x16x4_f32(false, af[s], false, bf0[s],
                                                 (short)0, c0, false, false);
      c1 = __builtin_amdgcn_wmma_f32_16x16x4_f32(false, af[s], false, bf1[s],
                                                 (short)0, c1, false, false);
    }

    if (c + 1 < nChunks) stage((c + 1) & 1);  // other buffer: no race with readers
    __syncthreads();
  }

  // ---- Epilogue: bias + hw tanh + store (C/D: vgpr r -> M=r | M=8+r) ----
  const int n0 = blockN + wx * 32 + l16;
  const int n1 = n0 + 16;
  const float bias0 = bias[n0];
  const float bias1 = bias[n1];
#pragma unroll
  for (int r = 0; r < 8; ++r) {
    const long m = blockM + wy * 16 + ((lane < 16) ? r : (8 + r));
    h_out[m * LATENT_ + n0] = fast_tanh(c0[r] + bias0);
    h_out[m * LATENT_ + n1] = fast_tanh(c1[r] + bias1);
  }
}

extern "C" void kernel_launch(void* const* d_in, const int* in_sizes, int n_in,
                              void* d_out, int out_size, void* d_ws, size_t ws_size,
                              hipStream_t stream) {
  const float* x    = (const float*)d_in[0];   // (B, T, DIM)
  const float* W    = (const float*)d_in[1];   // (DIM+LATENT, LATENT)
  const float* bias = (const float*)d_in[2];   // (LATENT)
  float* out = (float*)d_out;                  // (T, B, LATENT)

  dim3 grid(LATENT_ / 64, BATCH_ / 64);        // (16, 2)
  dim3 block(256);

  for (int t = 0; t < TSTEPS_; ++t) {
    const float* h_prev = (t == 0) ? out : (out + (long)(t - 1) * BATCH_ * LATENT_);
    float* h_out = out + (long)t * BATCH_ * LATENT_;
    const int kTotal = (t == 0) ? DIM_ : KTOT_;   // h0 == 0 -> skip Wh at t=0
    rnn_step_kernel<<<grid, block, 0, stream>>>(x, W, bias, h_prev, h_out, t, kTotal);
  }
}